// Model_2130303779110
// MI455X (gfx1250) — compile-verified
//
#include <hip/hip_runtime.h>
#include <math.h>

// ---------------------------------------------------------------------------
// MI455X (gfx1250) implementation of the S-Mamba forecaster.
// GEMMs use V_WMMA_F32_16X16X4_F32 (exact fp32 on the matrix pipe - the model
// is latency/L2 bound at 23.3 TB/s, so fp32 WMMA is the right precision) with
// double-buffered GLOBAL_LOAD_ASYNC_TO_LDS_B128 tile staging (ASYNCcnt).
// ---------------------------------------------------------------------------

typedef __attribute__((ext_vector_type(2))) float v2f;
typedef __attribute__((ext_vector_type(8))) float v8f;

#define TM 128
#define TN 64
#define TK 32
#define TKP (TK + 4)   // LDS row stride: keeps b128 stores 16B aligned, no bank conflicts

__device__ __forceinline__ v8f wmma_f32_4(v2f a, v2f b, v8f c) {
#if defined(__HIP_DEVICE_COMPILE__)
  return __builtin_amdgcn_wmma_f32_16x16x4_f32(false, a, false, b, (short)0, c, false, false);
#else
  (void)a; (void)b; return c;
#endif
}

// async copy of 16 bytes global -> LDS (per lane); tracked by ASYNCcnt
__device__ __forceinline__ void async_load16(unsigned lds_off, const float* gptr) {
#if defined(__HIP_DEVICE_COMPILE__)
  asm volatile("global_load_async_to_lds_b128 %0, %1, off"
               :: "v"(lds_off), "v"(gptr) : "memory");
#else
  (void)lds_off; (void)gptr;
#endif
}
__device__ __forceinline__ void wait_async0() {
#if defined(__HIP_DEVICE_COMPILE__)
  asm volatile("s_wait_asynccnt 0" ::: "memory");
#endif
}

// C[M,N] = A[M,K] * W[N,K]^T  (+ epilogue)
// EPI: 0 = store, 1 = store+bias, 3 = accumulate+bias+sigmoid, 4 = store+bias+softplus
template <int EPI>
__global__ __launch_bounds__(256) void gemm_wmma_kernel(
    const float* __restrict__ A, int lda, const float* __restrict__ W, int ldw,
    const float* __restrict__ bias, float* __restrict__ C, int ldc,
    int M, int N, int K) {
  __shared__ __align__(16) float As[2][TM][TKP];
  __shared__ __align__(16) float Ws[2][TN][TKP];

  const int tid  = threadIdx.x;
  const int lane = tid & 31;
  const int wave = tid >> 5;
  const int wm = wave >> 1;           // 0..3 : 32-row slab
  const int wn = wave & 1;            // 0..1 : 32-col slab
  const int rowBlk = blockIdx.y * TM;
  const int colBlk = blockIdx.x * TN;
  const int rowBase = rowBlk + wm * 32;
  const int colBase = colBlk + wn * 32;
  const int lrow  = lane & 15;
  const int khalf = (lane >> 4) << 1; // 0 or 2 (K-pair per lane half)

  v8f acc[2][2];
  #pragma unroll
  for (int i = 0; i < 2; ++i)
    #pragma unroll
    for (int j = 0; j < 2; ++j)
      #pragma unroll
      for (int r = 0; r < 8; ++r) acc[i][j][r] = 0.f;

  // stage one K-tile into LDS buffer `buf`: async b128 for fully in-range
  // chunks, synchronous guarded staging for K/N edge chunks.
  auto stage = [&](int buf, int k0) {
    const bool fullK = (k0 + TK) <= K;
    // A tile (128 x 32)
    for (int i = tid; i < TM * (TK / 4); i += 256) {
      int r = i >> 3, c = (i & 7) << 2;
      int gr = rowBlk + r, gc = k0 + c;
      if (gr < M && fullK) {
        async_load16((unsigned)(size_t)&As[buf][r][c], A + (size_t)gr * lda + gc);
      } else {
        float4 v = {0.f, 0.f, 0.f, 0.f};
        if (gr < M) {
          v.x = (gc + 0 < K) ? A[(size_t)gr * lda + gc + 0] : 0.f;
          v.y = (gc + 1 < K) ? A[(size_t)gr * lda + gc + 1] : 0.f;
          v.z = (gc + 2 < K) ? A[(size_t)gr * lda + gc + 2] : 0.f;
          v.w = (gc + 3 < K) ? A[(size_t)gr * lda + gc + 3] : 0.f;
        }
        *(float4*)&As[buf][r][c] = v;
      }
    }
    // W tile (64 x 32)
    for (int i = tid; i < TN * (TK / 4); i += 256) {
      int r = i >> 3, c = (i & 7) << 2;
      int gn = colBlk + r, gc = k0 + c;
      if (gn < N && fullK) {
        async_load16((unsigned)(size_t)&Ws[buf][r][c], W + (size_t)gn * ldw + gc);
      } else {
        float4 v = {0.f, 0.f, 0.f, 0.f};
        if (gn < N) {
          v.x = (gc + 0 < K) ? W[(size_t)gn * ldw + gc + 0] : 0.f;
          v.y = (gc + 1 < K) ? W[(size_t)gn * ldw + gc + 1] : 0.f;
          v.z = (gc + 2 < K) ? W[(size_t)gn * ldw + gc + 2] : 0.f;
          v.w = (gc + 3 < K) ? W[(size_t)gn * ldw + gc + 3] : 0.f;
        }
        *(float4*)&Ws[buf][r][c] = v;
      }
    }
  };

  int buf = 0;
  stage(0, 0);
  for (int k0 = 0; k0 < K; k0 += TK) {
    wait_async0();        // this wave's async tile loads complete
    __syncthreads();      // everyone's loads complete -> tile `buf` ready
    if (k0 + TK < K) stage(buf ^ 1, k0 + TK);  // overlap next fetch with math
    float (*Asb)[TKP] = As[buf];
    float (*Wsb)[TKP] = Ws[buf];
    #pragma unroll
    for (int kk = 0; kk < TK; kk += 4) {
      v2f a0 = *(const v2f*)&Asb[wm * 32 + lrow][kk + khalf];
      v2f a1 = *(const v2f*)&Asb[wm * 32 + 16 + lrow][kk + khalf];
      v2f b0 = *(const v2f*)&Wsb[wn * 32 + lrow][kk + khalf];
      v2f b1 = *(const v2f*)&Wsb[wn * 32 + 16 + lrow][kk + khalf];
      acc[0][0] = wmma_f32_4(a0, b0, acc[0][0]);
      acc[0][1] = wmma_f32_4(a0, b1, acc[0][1]);
      acc[1][0] = wmma_f32_4(a1, b0, acc[1][0]);
      acc[1][1] = wmma_f32_4(a1, b1, acc[1][1]);
    }
    buf ^= 1;
  }

  const int rsel = (lane >> 4) << 3;  // 0 or 8
  #pragma unroll
  for (int i = 0; i < 2; ++i) {
    #pragma unroll
    for (int j = 0; j < 2; ++j) {
      int col = colBase + j * 16 + lrow;
      float bv = 0.f;
      if ((EPI == 1 || EPI == 3 || EPI == 4) && col < N) bv = bias[col];
      #pragma unroll
      for (int r = 0; r < 8; ++r) {
        int row = rowBase + i * 16 + rsel + r;
        if (row < M && col < N) {
          float v = acc[i][j][r];
          if (EPI == 3) v += C[(size_t)row * ldc + col];
          if (EPI == 1 || EPI == 3 || EPI == 4) v += bv;
          if (EPI == 3) v = 1.f / (1.f + __expf(-v));
          if (EPI == 4) v = (v > 20.f) ? v : log1pf(expf(v));
          C[(size_t)row * ldc + col] = v;
        }
      }
    }
  }
}

// Selective scan: 1 thread per (b,d); h[16] in registers; B_t/C_t via LDS.
// Fuses output gating: y = (scan + xc*D) * silu(z)
__global__ __launch_bounds__(256) void scan_kernel(
    const float* __restrict__ xc, const float* __restrict__ dt,
    const float* __restrict__ dbc, int ldbc, int dr,
    const float* __restrict__ A_log, const float* __restrict__ Dp,
    const float* __restrict__ z, int ldz,
    float* __restrict__ y, int L, int di) {
  const int d = blockIdx.x * blockDim.x + threadIdx.x;
  const int b = blockIdx.y;
  __shared__ float sB[16], sC[16];
  float A[16], h[16];
  #pragma unroll
  for (int n = 0; n < 16; ++n) { A[n] = -expf(A_log[d * 16 + n]); h[n] = 0.f; }
  const float Dv = Dp[d];
  for (int t = 0; t < L; ++t) {
    const size_t row = (size_t)b * L + t;
    if (threadIdx.x < 32) {
      int n = threadIdx.x & 15;
      float v = dbc[row * ldbc + dr + ((threadIdx.x >> 4) << 4) + n];
      if (threadIdx.x < 16) sB[n] = v; else sC[n] = v;
    }
    __syncthreads();
    const float dtv = dt[row * di + d];
    const float xv  = xc[row * di + d];
    float accv = 0.f;
    #pragma unroll
    for (int n = 0; n < 16; ++n) {
      float da = expf(dtv * A[n]);
      h[n] = da * h[n] + dtv * sB[n] * xv;
      accv += h[n] * sC[n];
    }
    const float zv = z[row * ldz + d];
    y[row * di + d] = (accv + xv * Dv) * (zv / (1.f + expf(-zv)));
    __syncthreads();
  }
}

__global__ void conv_silu_kernel(const float* __restrict__ xz, int ldxz,
                                 const float* __restrict__ cw, const float* __restrict__ cb,
                                 float* __restrict__ xc, int L, int di, int k, int total) {
  int idx = blockIdx.x * blockDim.x + threadIdx.x;
  if (idx >= total) return;
  int d = idx % di, row = idx / di, t = row % L, b = row / L;
  float acc = cb[d];
  for (int i = 0; i < k; ++i) {
    int tt = t - (k - 1) + i;
    if (tt >= 0) acc += cw[d * k + i] * xz[(size_t)(b * L + tt) * ldxz + d];
  }
  xc[(size_t)row * di + d] = acc / (1.f + __expf(-acc));
}

__global__ void flip_kernel(const float* __restrict__ in, float* __restrict__ out,
                            int L, int dm, int total) {
  int idx = blockIdx.x * blockDim.x + threadIdx.x;
  if (idx >= total) return;
  int d = idx % dm, row = idx / dm, t = row % L, b = row / L;
  out[(size_t)(b * L + (L - 1 - t)) * dm + d] = in[idx];
}

__global__ void gate_combine_kernel(const float* __restrict__ g, const float* __restrict__ yf,
                                    const float* __restrict__ yb, float* __restrict__ out, int total) {
  int idx = blockIdx.x * blockDim.x + threadIdx.x;
  if (idx >= total) return;
  float zv = g[idx];
  out[idx] = zv * yf[idx] + (1.f - zv) * yb[idx];
}

__global__ void add_kernel(const float* __restrict__ a, const float* __restrict__ b,
                           float* __restrict__ out, int total) {
  int idx = blockIdx.x * blockDim.x + threadIdx.x;
  if (idx >= total) return;
  out[idx] = a[idx] + b[idx];
}

// MODE 1: c = p0+p1 ; MODE 2: c = p0*p1+p2. One block per row (D<=512); LDS
// staging makes in-place (out==p2) safe.
template <int MODE>
__global__ __launch_bounds__(256) void ln_kernel(
    const float* __restrict__ p0, const float* __restrict__ p1, const float* __restrict__ p2,
    const float* __restrict__ g, const float* __restrict__ be,
    float* __restrict__ out, int D) {
  __shared__ float srow[512];
  __shared__ float rs[256], rq[256];
  const int row = blockIdx.x;
  const size_t base = (size_t)row * D;
  float s = 0.f, q = 0.f;
  for (int i = threadIdx.x; i < D; i += 256) {
    float c = (MODE == 1) ? (p0[base + i] + p1[base + i])
                          : (p0[base + i] * p1[base + i] + p2[base + i]);
    srow[i] = c; s += c; q += c * c;
  }
  rs[threadIdx.x] = s; rq[threadIdx.x] = q; __syncthreads();
  for (int off = 128; off; off >>= 1) {
    if (threadIdx.x < off) { rs[threadIdx.x] += rs[threadIdx.x + off]; rq[threadIdx.x] += rq[threadIdx.x + off]; }
    __syncthreads();
  }
  float m = rs[0] / D;
  float var = rq[0] / D - m * m;
  float rstd = rsqrtf(var + 1e-5f);
  for (int i = threadIdx.x; i < D; i += 256)
    out[base + i] = g[i] * (srow[i] - m) * rstd + be[i];
}

__global__ void decomp_kernel(const float* __restrict__ x, float* __restrict__ s,
                              float* __restrict__ t, int D, int total) {
  int idx = blockIdx.x * blockDim.x + threadIdx.x;
  if (idx >= total) return;
  int col = idx % D, row = idx / D;
  const float* xr = x + (size_t)row * D;
  float sum = 0.f;
  #pragma unroll
  for (int j = -6; j <= 6; ++j) {
    int c = col + j; c = c < 0 ? 0 : (c >= D ? D - 1 : c);
    sum += xr[c];
  }
  float m = sum * (1.f / 13.f);
  s[idx] = xr[col] - m;
  t[idx] = m;
}

__global__ __launch_bounds__(256) void stats_kernel(const float* __restrict__ x,
                                                    float* __restrict__ mean,
                                                    float* __restrict__ stdev) {
  int row = blockIdx.x, b = row >> 7, c = row & 127;
  __shared__ float rs[256], rq[256];
  float s = 0.f, q = 0.f;
  for (int t = threadIdx.x; t < 512; t += 256) {
    float v = x[((size_t)b * 512 + t) * 128 + c];
    s += v; q += v * v;
  }
  rs[threadIdx.x] = s; rq[threadIdx.x] = q; __syncthreads();
  for (int off = 128; off; off >>= 1) {
    if (threadIdx.x < off) { rs[threadIdx.x] += rs[threadIdx.x + off]; rq[threadIdx.x] += rq[threadIdx.x + off]; }
    __syncthreads();
  }
  if (threadIdx.x == 0) {
    float m = rs[0] * (1.f / 512.f);
    float var = rq[0] * (1.f / 512.f) - m * m;
    mean[row] = m;
    stdev[row] = sqrtf(var + 1e-5f);
  }
}

__global__ void norm_transpose_kernel(const float* __restrict__ x, const float* __restrict__ mean,
                                      const float* __restrict__ stdev, float* __restrict__ out, int total) {
  int idx = blockIdx.x * blockDim.x + threadIdx.x;
  if (idx >= total) return;
  int t = idx % 512, row = idx / 512, b = row >> 7, c = row & 127;
  out[idx] = (x[((size_t)b * 512 + t) * 128 + c] - mean[row]) / stdev[row];
}

__global__ void denorm_kernel(const float* __restrict__ P, const float* __restrict__ mean,
                              const float* __restrict__ stdev, float* __restrict__ out, int total) {
  int idx = blockIdx.x * blockDim.x + threadIdx.x;
  if (idx >= total) return;
  int c = idx % 128, tp = (idx / 128) % 96, b = idx / (128 * 96);
  int row = b * 128 + c;
  out[idx] = P[(size_t)row * 96 + tp] * stdev[row] + mean[row];
}

// (B,R,C) -> (B,C,R)
__global__ void transpose_kernel(const float* __restrict__ in, float* __restrict__ out, int R, int C) {
  __shared__ float tile[32][33];
  int b = blockIdx.z;
  int r0 = blockIdx.y * 32, c0 = blockIdx.x * 32;
  const float* ib = in + (size_t)b * R * C;
  float* ob = out + (size_t)b * R * C;
  for (int j = threadIdx.y; j < 32; j += 8) {
    int r = r0 + j, c = c0 + threadIdx.x;
    if (r < R && c < C) tile[j][threadIdx.x] = ib[(size_t)r * C + c];
  }
  __syncthreads();
  for (int j = threadIdx.y; j < 32; j += 8) {
    int r = c0 + j, c = r0 + threadIdx.x;
    if (r < C && c < R) ob[(size_t)r * R + c] = tile[threadIdx.x][j];
  }
}

// ---------------------------------------------------------------------------
// Host side
// ---------------------------------------------------------------------------

struct MambaW { const float *A_log, *D, *conv_b, *conv_w, *dt_b, *dt_w, *in_w, *out_w, *xp_w; };
struct BiW    { MambaW mb, mf; const float *gate_b, *gate_w, *out_b, *out_w; };
struct EncW   { BiW cm; const float *n1_b, *n1_g, *n2_b, *n2_g; MambaW tm; };

struct WS {
  float *MEANS, *STDEV, *TRBUF, *XBUF, *SBUF, *TBUF, *Y0, *YF, *YB, *XFLIP,
        *GTMP, *GBUF, *BOUT, *MXZ, *MXC, *MDBC, *MDT, *MY, *PBUF;
};

static MambaW read_mamba(void* const* d, int& i) {
  MambaW w;
  w.A_log = (const float*)d[i++]; w.D     = (const float*)d[i++];
  w.conv_b= (const float*)d[i++]; w.conv_w= (const float*)d[i++];
  w.dt_b  = (const float*)d[i++]; w.dt_w  = (const float*)d[i++];
  w.in_w  = (const float*)d[i++]; w.out_w = (const float*)d[i++];
  w.xp_w  = (const float*)d[i++];
  return w;
}
static BiW read_bi(void* const* d, int& i) {
  BiW w; w.mb = read_mamba(d, i); w.mf = read_mamba(d, i);
  w.gate_b = (const float*)d[i++]; w.gate_w = (const float*)d[i++];
  w.out_b  = (const float*)d[i++]; w.out_w  = (const float*)d[i++];
  return w;
}
static EncW read_enc(void* const* d, int& i) {
  EncW e; e.cm = read_bi(d, i);
  e.n1_b = (const float*)d[i++]; e.n1_g = (const float*)d[i++];
  e.n2_b = (const float*)d[i++]; e.n2_g = (const float*)d[i++];
  e.tm = read_mamba(d, i);
  return e;
}

static void gemm(int epi, const float* A, int lda, const float* W, int ldw,
                 const float* bias, float* C, int ldc, int M, int N, int K,
                 hipStream_t s) {
  dim3 grid((N + TN - 1) / TN, (M + TM - 1) / TM), blk(256);
  switch (epi) {
    case 0: gemm_wmma_kernel<0><<<grid, blk, 0, s>>>(A, lda, W, ldw, bias, C, ldc, M, N, K); break;
    case 1: gemm_wmma_kernel<1><<<grid, blk, 0, s>>>(A, lda, W, ldw, bias, C, ldc, M, N, K); break;
    case 3: gemm_wmma_kernel<3><<<grid, blk, 0, s>>>(A, lda, W, ldw, bias, C, ldc, M, N, K); break;
    case 4: gemm_wmma_kernel<4><<<grid, blk, 0, s>>>(A, lda, W, ldw, bias, C, ldc, M, N, K); break;
  }
}

static void run_mamba(const MambaW& m, const float* x, float* out,
                      int Bb, int L, int dm, int di, int dr, int k,
                      WS& w, hipStream_t s) {
  const int T = Bb * L;
  // xz = x @ in_w^T   (T, 2*di)
  gemm(0, x, dm, m.in_w, dm, nullptr, w.MXZ, 2 * di, T, 2 * di, dm, s);
  // xc = silu(causal_conv(xin)+b)
  int tot = T * di;
  conv_silu_kernel<<<(tot + 255) / 256, 256, 0, s>>>(w.MXZ, 2 * di, m.conv_w, m.conv_b, w.MXC, L, di, k, tot);
  // dbc = xc @ xp_w^T  (T, dr+32)
  gemm(0, w.MXC, di, m.xp_w, di, nullptr, w.MDBC, dr + 32, T, dr + 32, di, s);
  // dt = softplus(dbc[:, :dr] @ dt_w^T + dt_b)  (T, di)
  gemm(4, w.MDBC, dr + 32, m.dt_w, dr, m.dt_b, w.MDT, di, T, di, dr, s);
  // selective scan + output gating -> MY (T, di)
  dim3 sg(di / 256, Bb);
  scan_kernel<<<sg, 256, 0, s>>>(w.MXC, w.MDT, w.MDBC, dr + 32, dr, m.A_log, m.D,
                                 w.MXZ + di, 2 * di, w.MY, L, di);
  // out = MY @ out_w^T  (T, dm)
  gemm(0, w.MY, di, m.out_w, di, nullptr, out, dm, T, dm, di, s);
}

static void run_bimamba(const BiW& bw, const float* x, float* out,
                        int Bb, int L, int dm, int di, int dr, int k,
                        WS& w, hipStream_t s) {
  const int T = Bb * L, tot = T * dm;
  run_mamba(bw.mf, x, w.YF, Bb, L, dm, di, dr, k, w, s);
  flip_kernel<<<(tot + 255) / 256, 256, 0, s>>>(x, w.XFLIP, L, dm, tot);
  run_mamba(bw.mb, w.XFLIP, w.GTMP, Bb, L, dm, di, dr, k, w, s);
  flip_kernel<<<(tot + 255) / 256, 256, 0, s>>>(w.GTMP, w.YB, L, dm, tot);
  // gate = sigmoid(yf @ W1^T + yb @ W2^T + b)  (split-K of the concat)
  gemm(0, w.YF, dm, bw.gate_w, 2 * dm, nullptr, w.GBUF, dm, T, dm, dm, s);
  gemm(3, w.YB, dm, bw.gate_w + dm, 2 * dm, bw.gate_b, w.GBUF, dm, T, dm, dm, s);
  gate_combine_kernel<<<(tot + 255) / 256, 256, 0, s>>>(w.GBUF, w.YF, w.YB, w.GTMP, tot);
  gemm(1, w.GTMP, dm, bw.out_w, dm, bw.out_b, out, dm, T, dm, dm, s);
}

static void run_encoder(const EncW& e, float* x, WS& w, hipStream_t s) {
  // channel bimamba: B=8, L=128, dm=512, di=1024, dr=32, k=1
  run_bimamba(e.cm, x, w.BOUT, 8, 128, 512, 1024, 32, 1, w, s);
  ln_kernel<1><<<1024, 256, 0, s>>>(w.BOUT, x, nullptr, e.n1_g, e.n1_b, w.Y0, 512);
  // time mamba on transposed (B,512,128): dm=128, di=256, dr=8, k=4
  { dim3 g(512 / 32, 128 / 32, 8), b(32, 8);
    transpose_kernel<<<g, b, 0, s>>>(w.Y0, w.TRBUF, 128, 512); }
  run_mamba(e.tm, w.TRBUF, w.YF, 8, 512, 128, 256, 8, 4, w, s);
  { dim3 g(128 / 32, 512 / 32, 8), b(32, 8);
    transpose_kernel<<<g, b, 0, s>>>(w.YF, w.YB, 512, 128); }
  // x = LN(y1*y0 + x)  (in place; ln_kernel stages through LDS)
  ln_kernel<2><<<1024, 256, 0, s>>>(w.YB, w.Y0, x, e.n2_g, e.n2_b, x, 512);
}

extern "C" void kernel_launch(void* const* d_in, const int* in_sizes, int n_in,
                              void* d_out, int out_size, void* d_ws, size_t ws_size,
                              hipStream_t stream) {
  (void)in_sizes; (void)n_in; (void)out_size; (void)ws_size;
  const float* x_enc = (const float*)d_in[0];

  // params leaves in jax pytree order (dict keys sorted): emb, pre, proj, seasonal, trend
  int i = 4;
  const float* emb_b = (const float*)d_in[i++];
  const float* emb_w = (const float*)d_in[i++];
  BiW pre = read_bi(d_in, i);
  const float* proj_b = (const float*)d_in[i++];
  const float* proj_w = (const float*)d_in[i++];
  EncW seas0 = read_enc(d_in, i), seas1 = read_enc(d_in, i);
  EncW tren0 = read_enc(d_in, i), tren1 = read_enc(d_in, i);

  float* f = (float*)d_ws;
  WS w; size_t off = 0;
  auto alloc = [&](size_t n) { float* p = f + off; off += n; return p; };
  w.MEANS = alloc(1024);     w.STDEV = alloc(1024);
  w.TRBUF = alloc(524288);   w.XBUF  = alloc(524288);
  w.SBUF  = alloc(524288);   w.TBUF  = alloc(524288);
  w.Y0    = alloc(524288);   w.YF    = alloc(524288);
  w.YB    = alloc(524288);   w.XFLIP = alloc(524288);
  w.GTMP  = alloc(524288);   w.GBUF  = alloc(524288);
  w.BOUT  = alloc(524288);
  w.MXZ   = alloc(2097152);  w.MXC   = alloc(1048576);
  w.MDBC  = alloc(163840);   w.MDT   = alloc(1048576);
  w.MY    = alloc(1048576);  w.PBUF  = alloc(98304);

  // instance norm stats over seq dim
  stats_kernel<<<1024, 256, 0, stream>>>(x_enc, w.MEANS, w.STDEV);
  // normalized + channel-transposed tokens (1024, 512)
  norm_transpose_kernel<<<(524288 + 255) / 256, 256, 0, stream>>>(x_enc, w.MEANS, w.STDEV, w.TRBUF, 524288);
  // embedding: (1024,512) @ emb_w^T + b
  gemm(1, w.TRBUF, 512, emb_w, 512, emb_b, w.XBUF, 512, 1024, 512, 512, stream);
  // pre bimamba
  run_bimamba(pre, w.XBUF, w.Y0, 8, 128, 512, 1024, 32, 1, w, stream);
  // seasonal/trend decomposition (13-tap moving average on feature dim)
  decomp_kernel<<<(524288 + 255) / 256, 256, 0, stream>>>(w.Y0, w.SBUF, w.TBUF, 512, 524288);
  // encoder stacks
  run_encoder(seas0, w.SBUF, w, stream);
  run_encoder(seas1, w.SBUF, w, stream);
  run_encoder(tren0, w.TBUF, w, stream);
  run_encoder(tren1, w.TBUF, w, stream);
  // merge + projection to PRED_LEN
  add_kernel<<<(524288 + 255) / 256, 256, 0, stream>>>(w.SBUF, w.TBUF, w.XBUF, 524288);
  gemm(1, w.XBUF, 512, proj_w, 512, proj_b, w.PBUF, 96, 1024, 96, 512, stream);
  // de-normalize + final transpose into d_out (8, 96, 128)
  denorm_kernel<<<(98304 + 255) / 256, 256, 0, stream>>>(w.PBUF, w.MEANS, w.STDEV, (float*)d_out, 98304);
}